// TraceableCausalSelfAttention_66709432041838
// MI455X (gfx1250) — compile-verified
//
#include <hip/hip_runtime.h>
#include <math.h>

// ---------------------------------------------------------------------------
// Problem constants (from reference): B=16, S=4096, H=16, D=128, d_model=2048
// ---------------------------------------------------------------------------
#define D_MODEL   2048
#define HD        2048            // H*D
#define QKV_F     6144            // 3*H*D
#define BATCH     16
#define SEQ       4096
#define HEADS     16
#define HDIM      128
#define CACHE_HALF 134217728      // B*S*H*D floats (one of k/v halves)
#define CACHE_ELEMS 268435456     // 2 * CACHE_HALF

typedef __attribute__((ext_vector_type(2))) float v2f;
typedef __attribute__((ext_vector_type(8))) float v8f;

// ---------------------------------------------------------------------------
// GEMM with M=16 using V_WMMA_F32_16X16X4_F32.
//   C[16][N] = A[16][K] * B^T   where Bm is stored [N][K] row-major
// One wave computes one 16x16 C tile, looping K in steps of 4.
// Wave32 VGPR layouts (ISA 7.12.2):
//   A 16x4 : lane L, vgpr v -> M = L%16, K = v + 2*(L/16)
//   B 4x16 : lane L, vgpr v -> K = v + 2*(L/16), N = L%16
//   C 16x16: lane L, vgpr v -> M = v + 8*(L/16), N = L%16
// ---------------------------------------------------------------------------
__global__ __launch_bounds__(256)
void gemm16_wmma_kernel(const float* __restrict__ A,   // [16][K]
                        const float* __restrict__ Bm,  // [N][K]
                        float* __restrict__ Cm,        // [16][N]
                        int K, int N) {
  const int lane = threadIdx.x & 31;
  const int wid  = threadIdx.x >> 5;
  const int tile = blockIdx.x * 8 + wid;     // one 16-wide N tile per wave
  const int n0   = tile * 16;
  if (n0 >= N) return;                       // wave-uniform, EXEC stays all-1s

  const int half = lane >> 4;                // 0: lanes 0-15, 1: lanes 16-31
  const int l    = lane & 15;

  const float* arow = A  + (size_t)l * K + 2 * half;          // M=l, K base
  const float* brow = Bm + (size_t)(n0 + l) * K + 2 * half;   // N=n0+l, K base

  v8f c = {0.f, 0.f, 0.f, 0.f, 0.f, 0.f, 0.f, 0.f};
#pragma unroll 4
  for (int k = 0; k < K; k += 4) {
    v2f a = { arow[k], arow[k + 1] };        // A[M=l][K = k+2*half (+1)]
    v2f b = { brow[k], brow[k + 1] };        // B[K = k+2*half (+1)][N=n0+l]
    // 8 args: (neg_a, A, neg_b, B, c_mod, C, reuse_a, reuse_b)
    c = __builtin_amdgcn_wmma_f32_16x16x4_f32(false, a, false, b,
                                              (short)0, c, false, false);
  }
#pragma unroll
  for (int v = 0; v < 8; ++v) {
    const int m = v + 8 * half;              // batch row
    Cm[(size_t)m * N + n0 + l] = c[v];
  }
}

// ---------------------------------------------------------------------------
// Overwrite the updated cache rows (k,v at position) and write new_position.
// 2 * 16 * 2048 = 65536 floats total.
// ---------------------------------------------------------------------------
__global__ __launch_bounds__(256)
void cache_update_kernel(const float* __restrict__ qkv,       // [16][6144]
                         const float* __restrict__ position,
                         float* __restrict__ out_cache,       // [2][16][4096][16][128]
                         float* __restrict__ new_position_out) {
  const int pos = (int)position[0];
  const int idx = blockIdx.x * 256 + threadIdx.x;             // 0 .. 65535
  const int kvsel = idx >> 15;            // 0 = k, 1 = v
  const int rem   = idx & 32767;
  const int b     = rem >> 11;
  const int c     = rem & 2047;           // h*128 + d
  const size_t dst = (size_t)kvsel * CACHE_HALF
                   + ((size_t)b * SEQ + (size_t)pos) * HD + c;
  out_cache[dst] = qkv[(size_t)b * QKV_F + (size_t)(1 + kvsel) * HD + c];
  if (idx == 0) new_position_out[0] = position[0] + 1.0f;
}

// ---------------------------------------------------------------------------
// Attention: one block per (b,h); 16 waves; one wave owns one K/V row at a
// time (32 lanes x float4 = full 512B row, perfectly coalesced).
// Causal mask -> only s in [0, pos]. Row s==pos comes from fresh qkv, the
// rest from the ORIGINAL cache (the output copy is stream-ordered before this kernel).
// ---------------------------------------------------------------------------
#define AW 16   // waves per block (512 threads)

__global__ __launch_bounds__(512)
void attention_kernel(const float* __restrict__ kv_cache,  // [2][16][4096][16][128]
                      const float* __restrict__ position,
                      const float* __restrict__ qkv,       // [16][6144]
                      float* __restrict__ attn_out) {      // [16][2048]
  __shared__ float sScore[SEQ];
  __shared__ float sRed[AW];
  __shared__ float sStat[2];               // [0]=max, [1]=sum
  __shared__ float sAcc[AW][HDIM];

  const int tid  = threadIdx.x;
  const int lane = tid & 31;
  const int wid  = tid >> 5;
  const int b    = blockIdx.x >> 4;
  const int h    = blockIdx.x & 15;
  const int pos  = (int)position[0];
  const int n    = pos + 1;
  const float scale = 0.08838834764831845f;   // 128^-0.5

  // q for this (b,h): lane holds d = 4*lane .. 4*lane+3
  const float* qptr = qkv + (size_t)b * QKV_F + (size_t)h * HDIM;
  const float4 qv = *(const float4*)(qptr + lane * 4);

  const float* kbase = kv_cache + ((size_t)b * SEQ * HEADS + h) * HDIM;
  const float* knew  = qkv + (size_t)b * QKV_F + HD + (size_t)h * HDIM;

  // ---- pass 1: scores ----
  for (int s = wid; s < n; s += AW) {
    const float* kr = (s == pos) ? knew : (kbase + (size_t)s * HD);
    if (s + AW < pos)
      __builtin_prefetch(kbase + (size_t)(s + AW) * HD + lane * 4, 0, 3);
    const float4 k4 = *(const float4*)(kr + lane * 4);
    float p = qv.x * k4.x + qv.y * k4.y + qv.z * k4.z + qv.w * k4.w;
#pragma unroll
    for (int off = 16; off > 0; off >>= 1) p += __shfl_down(p, off, 32);
    if (lane == 0) sScore[s] = p * scale;
  }
  __syncthreads();

  // ---- softmax statistics ----
  float m = -INFINITY;
  for (int i = tid; i < n; i += 512) m = fmaxf(m, sScore[i]);
#pragma unroll
  for (int off = 16; off > 0; off >>= 1) m = fmaxf(m, __shfl_down(m, off, 32));
  if (lane == 0) sRed[wid] = m;
  __syncthreads();
  if (tid == 0) {
    float mm = sRed[0];
    for (int i = 1; i < AW; ++i) mm = fmaxf(mm, sRed[i]);
    sStat[0] = mm;
  }
  __syncthreads();
  const float M = sStat[0];

  float lsum = 0.f;
  for (int i = tid; i < n; i += 512) {
    const float e = __expf(sScore[i] - M);
    sScore[i] = e;
    lsum += e;
  }
#pragma unroll
  for (int off = 16; off > 0; off >>= 1) lsum += __shfl_down(lsum, off, 32);
  if (lane == 0) sRed[wid] = lsum;
  __syncthreads();
  if (tid == 0) {
    float ss = 0.f;
    for (int i = 0; i < AW; ++i) ss += sRed[i];
    sStat[1] = ss;
  }
  __syncthreads();
  const float inv = 1.0f / sStat[1];

  // ---- pass 2: weighted V sum ----
  const float* vbase = kv_cache + (size_t)CACHE_HALF
                     + ((size_t)b * SEQ * HEADS + h) * HDIM;
  const float* vnew  = qkv + (size_t)b * QKV_F + 2 * HD + (size_t)h * HDIM;
  float4 acc = {0.f, 0.f, 0.f, 0.f};
  for (int s = wid; s < n; s += AW) {
    const float* vr = (s == pos) ? vnew : (vbase + (size_t)s * HD);
    if (s + AW < pos)
      __builtin_prefetch(vbase + (size_t)(s + AW) * HD + lane * 4, 0, 3);
    const float4 v4 = *(const float4*)(vr + lane * 4);
    const float p = sScore[s];
    acc.x = fmaf(p, v4.x, acc.x);
    acc.y = fmaf(p, v4.y, acc.y);
    acc.z = fmaf(p, v4.z, acc.z);
    acc.w = fmaf(p, v4.w, acc.w);
  }
  *(float4*)(&sAcc[wid][lane * 4]) = acc;
  __syncthreads();

  if (tid < HDIM) {
    float r = 0.f;
#pragma unroll
    for (int w = 0; w < AW; ++w) r += sAcc[w][tid];
    attn_out[(size_t)b * HD + (size_t)h * HDIM + tid] = r * inv;
  }
}

// ---------------------------------------------------------------------------
// Launch
// ---------------------------------------------------------------------------
extern "C" void kernel_launch(void* const* d_in, const int* in_sizes, int n_in,
                              void* d_out, int out_size, void* d_ws, size_t ws_size,
                              hipStream_t stream) {
  const float* x        = (const float*)d_in[0];  // [16][1][2048]
  const float* kv_cache = (const float*)d_in[1];  // [2][16][4096][16][128]
  const float* position = (const float*)d_in[2];  // [1]
  const float* W_qkv    = (const float*)d_in[3];  // [6144][2048]
  const float* W_o      = (const float*)d_in[4];  // [2048][2048] (m, f)

  float* out       = (float*)d_out;               // [16][2048]
  float* out_cache = out + (size_t)BATCH * HD;    // 1 GiB cache copy
  float* out_pos   = out_cache + (size_t)CACHE_ELEMS;

  float* qkv_ws  = (float*)d_ws;                          // [16][6144]
  float* attn_ws = qkv_ws + (size_t)BATCH * QKV_F;        // [16][2048]

  // 1) QKV projection: qkv[b][f] = sum_m x[b][m] * W_qkv[f][m]   (WMMA f32)
  gemm16_wmma_kernel<<<48, 256, 0, stream>>>(x, W_qkv, qkv_ws, D_MODEL, QKV_F);

  // 2) Functional cache copy (bandwidth-dominant: 1 GiB -> 1 GiB, driver path)
  hipMemcpyAsync(out_cache, kv_cache, (size_t)CACHE_ELEMS * sizeof(float),
                 hipMemcpyDeviceToDevice, stream);

  // 3) Scatter the new k/v rows at `pos` + write new_position
  cache_update_kernel<<<256, 256, 0, stream>>>(qkv_ws, position, out_cache, out_pos);

  // 4) Attention over s in [0, pos] (reads original cache + fresh k/v row)
  attention_kernel<<<BATCH * HEADS, 512, 0, stream>>>(kv_cache, position,
                                                      qkv_ws, attn_ws);

  // 5) Output projection: out[b][m] = sum_f attn[b][f] * W_o[m][f]  (WMMA f32)
  gemm16_wmma_kernel<<<16, 256, 0, stream>>>(attn_ws, W_o, out, HD, D_MODEL);
}